// Net_2207613190837
// MI455X (gfx1250) — compile-verified
//
#include <hip/hip_runtime.h>
#include <math.h>

typedef __attribute__((ext_vector_type(2))) float v2f;
typedef __attribute__((ext_vector_type(8))) float v8f;

// ---------------------------------------------------------------------------
// Utility fills
// ---------------------------------------------------------------------------
__global__ void fill_f32(float* __restrict__ p, float v, long long n) {
    long long i = (long long)blockIdx.x * blockDim.x + threadIdx.x;
    if (i < n) p[i] = v;
}

// Build zero-padded W2: [16 x 11] -> [16 x 16] (cols 11..15 = 0)
__global__ void pad_w2(const float* __restrict__ W2, float* __restrict__ w2p) {
    int t = threadIdx.x;            // 256 threads
    int k = t >> 4, c = t & 15;
    w2p[t] = (c < 11) ? W2[k * 11 + c] : 0.0f;
}

// ---------------------------------------------------------------------------
// Degree / normalization
// ---------------------------------------------------------------------------
__global__ void degree_kernel(const int* __restrict__ dst, float* __restrict__ deg,
                              long long E) {
    long long e = (long long)blockIdx.x * blockDim.x + threadIdx.x;
    if (e < E) atomicAdd(&deg[dst[e]], 1.0f);
}

__global__ void rsqrt_inplace(float* __restrict__ deg, int N) {
    int i = blockIdx.x * blockDim.x + threadIdx.x;
    if (i < N) deg[i] = rsqrtf(deg[i]);   // deg >= 1 due to self loop
}

// ---------------------------------------------------------------------------
// WMMA GEMM: [N,128] x [128,16] -> [N,16]   (fp32, V_WMMA_F32_16X16X4_F32)
// One wave per 16-row tile. N must be a multiple of 16 (100000 = 6250*16).
// A 16x4 fragment: lane l holds rows m=l&15, K = 2*(l>>4)+{0,1}  -> float2.
// B  4x16 fragment: lane l holds col n=l&15, K = 2*(l>>4)+{0,1}.
// C/D 16x16: VGPR v holds M = v + 8*(l>>4), N = l&15.
// ---------------------------------------------------------------------------
__global__ void gemm1_wmma(const float* __restrict__ X,
                           const float* __restrict__ W1,
                           float* __restrict__ out, int numTiles) {
    int wave = blockIdx.x * (blockDim.x >> 5) + (threadIdx.x >> 5);
    if (wave >= numTiles) return;                 // wave-uniform branch
    int lane = threadIdx.x & 31;
    int half = lane >> 4;                         // 0 or 1
    int sub  = lane & 15;                         // row (A) / col (B,D)
    int kb   = half * 2;

    const float* xrow = X + ((long long)(wave * 16 + sub)) * 128;
    v8f c = {0.f, 0.f, 0.f, 0.f, 0.f, 0.f, 0.f, 0.f};

#pragma unroll 4
    for (int k = 0; k < 128; k += 4) {
        v2f a = *(const v2f*)(xrow + k + kb);     // contiguous, 8B aligned
        v2f b;
        b.x = W1[(k + kb + 0) * 16 + sub];
        b.y = W1[(k + kb + 1) * 16 + sub];
        c = __builtin_amdgcn_wmma_f32_16x16x4_f32(false, a, false, b,
                                                  (short)0, c, false, false);
    }

    long long rb = (long long)wave * 16;
#pragma unroll
    for (int v = 0; v < 8; ++v)
        out[(rb + v + half * 8) * 16 + sub] = c[v];
}

// WMMA GEMM: [N,16] x [16,16] -> [N,16]  (K = 16, 4 wmma steps)
__global__ void gemm2_wmma(const float* __restrict__ H,
                           const float* __restrict__ W2p,
                           float* __restrict__ out, int numTiles) {
    int wave = blockIdx.x * (blockDim.x >> 5) + (threadIdx.x >> 5);
    if (wave >= numTiles) return;
    int lane = threadIdx.x & 31;
    int half = lane >> 4;
    int sub  = lane & 15;
    int kb   = half * 2;

    const float* hrow = H + ((long long)(wave * 16 + sub)) * 16;
    v8f c = {0.f, 0.f, 0.f, 0.f, 0.f, 0.f, 0.f, 0.f};

#pragma unroll
    for (int k = 0; k < 16; k += 4) {
        v2f a = *(const v2f*)(hrow + k + kb);
        v2f b;
        b.x = W2p[(k + kb + 0) * 16 + sub];
        b.y = W2p[(k + kb + 1) * 16 + sub];
        c = __builtin_amdgcn_wmma_f32_16x16x4_f32(false, a, false, b,
                                                  (short)0, c, false, false);
    }

    long long rb = (long long)wave * 16;
#pragma unroll
    for (int v = 0; v < 8; ++v)
        out[(rb + v + half * 8) * 16 + sub] = c[v];
}

// ---------------------------------------------------------------------------
// Edge message passing: 16 lanes per edge (feature j = lane&15).
// Gather xw[src*16+j] (coalesced 64B), scale by dinv[src]*dinv[dst],
// atomic scatter into acc[dst*16+j]. Works for both layers (layer2 padded
// columns are exact zeros so the extra atomics add 0).
// ---------------------------------------------------------------------------
__global__ void msg16(const float* __restrict__ xw, const float* __restrict__ dinv,
                      const int* __restrict__ src, const int* __restrict__ dst,
                      float* __restrict__ acc, long long E) {
    long long t = (long long)blockIdx.x * blockDim.x + threadIdx.x;
    long long e = t >> 4;
    int j = (int)(t & 15);
    if (e >= E) return;
    int s = src[e];
    int d = dst[e];
    float norm = dinv[s] * dinv[d];
    atomicAdd(&acc[(long long)d * 16 + j], xw[(long long)s * 16 + j] * norm);
}

// Layer-1 epilogue: self loop + bias + ReLU, in place (acc1 -> h1)
__global__ void selfloop_relu(const float* __restrict__ xw,
                              float* __restrict__ acc,
                              const float* __restrict__ dinv,
                              const float* __restrict__ b1, long long n16) {
    long long t = (long long)blockIdx.x * blockDim.x + threadIdx.x;
    if (t >= n16) return;
    long long i = t >> 4;
    int j = (int)(t & 15);
    float di = dinv[i];
    float h = acc[t] + xw[t] * di * di + b1[j];
    acc[t] = fmaxf(h, 0.0f);
}

// Layer-2 epilogue: self loop + bias + log_softmax over 11 classes
__global__ void finalize(const float* __restrict__ acc2,
                         const float* __restrict__ xw2,
                         const float* __restrict__ dinv,
                         const float* __restrict__ b2,
                         float* __restrict__ out, int N) {
    int i = blockIdx.x * blockDim.x + threadIdx.x;
    if (i >= N) return;
    float di = dinv[i];
    float sl = di * di;
    float z[11];
    float m = -3.402823466e+38f;
#pragma unroll
    for (int c = 0; c < 11; ++c) {
        float v = acc2[(long long)i * 16 + c] + xw2[(long long)i * 16 + c] * sl + b2[c];
        z[c] = v;
        m = fmaxf(m, v);
    }
    float ssum = 0.0f;
#pragma unroll
    for (int c = 0; c < 11; ++c) ssum += expf(z[c] - m);
    float lse = m + logf(ssum);
#pragma unroll
    for (int c = 0; c < 11; ++c) out[(long long)i * 11 + c] = z[c] - lse;
}

// ---------------------------------------------------------------------------
// Launch
// ---------------------------------------------------------------------------
extern "C" void kernel_launch(void* const* d_in, const int* in_sizes, int n_in,
                              void* d_out, int out_size, void* d_ws, size_t ws_size,
                              hipStream_t stream) {
    const float* x  = (const float*)d_in[0];   // [N,128]
    const int*   ei = (const int*)d_in[1];     // [2,E]
    const float* W1 = (const float*)d_in[2];   // [128,16]
    const float* b1 = (const float*)d_in[3];   // [16]
    const float* W2 = (const float*)d_in[4];   // [16,11]
    const float* b2 = (const float*)d_in[5];   // [11]
    float* out = (float*)d_out;                // [N,11]

    const int       N = in_sizes[0] / 128;     // 100000
    const long long E = (long long)in_sizes[1] / 2;
    const int*   src = ei;
    const int*   dst = ei + E;

    // Workspace layout (floats)
    float* ws   = (float*)d_ws;
    float* deg  = ws;                                  // N   (becomes dinv)
    float* xw1  = deg  + N;                            // 16N
    float* acc1 = xw1  + (long long)16 * N;            // 16N (becomes h1)
    float* xw2  = acc1 + (long long)16 * N;            // 16N
    float* acc2 = xw2  + (long long)16 * N;            // 16N
    float* w2p  = acc2 + (long long)16 * N;            // 256

    const int B = 256;
    const long long n16 = (long long)16 * N;
    const long long msgThreads = E * 16;

    // 1. init
    fill_f32<<<(int)((N + B - 1) / B), B, 0, stream>>>(deg, 1.0f, N);
    fill_f32<<<(int)((n16 + B - 1) / B), B, 0, stream>>>(acc1, 0.0f, n16);
    fill_f32<<<(int)((n16 + B - 1) / B), B, 0, stream>>>(acc2, 0.0f, n16);
    pad_w2<<<1, 256, 0, stream>>>(W2, w2p);

    // 2. degree + dinv
    degree_kernel<<<(int)((E + B - 1) / B), B, 0, stream>>>(dst, deg, E);
    rsqrt_inplace<<<(N + B - 1) / B, B, 0, stream>>>(deg, N);

    // 3. GEMM1 (WMMA): xw1 = x @ W1
    int tiles = N / 16;                                // N divisible by 16
    int wavesPerBlock = B / 32;
    int gemmBlocks = (tiles + wavesPerBlock - 1) / wavesPerBlock;
    gemm1_wmma<<<gemmBlocks, B, 0, stream>>>(x, W1, xw1, tiles);

    // 4. layer-1 message passing
    msg16<<<(int)((msgThreads + B - 1) / B), B, 0, stream>>>(xw1, deg, src, dst, acc1, E);

    // 5. self loop + bias + ReLU (acc1 -> h1)
    selfloop_relu<<<(int)((n16 + B - 1) / B), B, 0, stream>>>(xw1, acc1, deg, b1, n16);

    // 6. GEMM2 (WMMA): xw2 = h1 @ W2p
    gemm2_wmma<<<gemmBlocks, B, 0, stream>>>(acc1, w2p, xw2, tiles);

    // 7. layer-2 message passing
    msg16<<<(int)((msgThreads + B - 1) / B), B, 0, stream>>>(xw2, deg, src, dst, acc2, E);

    // 8. self loop + bias + log_softmax
    finalize<<<(N + B - 1) / B, B, 0, stream>>>(acc2, xw2, deg, b2, out, N);
}